// FastSoftMax_86363202388360
// MI455X (gfx1250) — compile-verified
//
#include <hip/hip_runtime.h>
#include <hip/hip_bf16.h>
#include <stdint.h>

#define NTHREADS 256
#define NWAVES 8

// ---- gfx1250 async global->LDS copy helpers -------------------------------
// Builtin prototype (from hipcc diagnostic):
//   void __builtin_amdgcn_global_load_async_to_lds_b128(
//       v4i __device__* gsrc, v4i __shared__* ldst, imm int offset, imm int cpol)
typedef int v4i __attribute__((__vector_size__(4 * sizeof(int))));
typedef __attribute__((address_space(1))) v4i GV4;   // global (AS1)
typedef __attribute__((address_space(3))) v4i LV4;   // LDS (AS3)

__device__ __forceinline__ void async_copy_b128(const float* gsrc, float* ldst) {
  // Low 32 bits of a flat shared pointer are the LDS byte offset (ISA §10.2).
  uint32_t loff = (uint32_t)(uintptr_t)ldst;
#if __has_builtin(__builtin_amdgcn_global_load_async_to_lds_b128)
  __builtin_amdgcn_global_load_async_to_lds_b128(
      (GV4*)(uintptr_t)gsrc, (LV4*)(uintptr_t)loff, 0, 0);
#else
  asm volatile("global_load_async_to_lds_b128 %0, %1, off"
               :
               : "v"(loff), "v"(gsrc)
               : "memory");
#endif
}

__device__ __forceinline__ void wait_async0() {
#if __has_builtin(__builtin_amdgcn_s_wait_asynccnt)
  __builtin_amdgcn_s_wait_asynccnt(0);
#else
  asm volatile("s_wait_asynccnt 0" ::: "memory");
#endif
}

// ---- wave32 butterfly reductions ------------------------------------------
__device__ __forceinline__ float wave_max(float v) {
#pragma unroll
  for (int d = 16; d >= 1; d >>= 1) v = fmaxf(v, __shfl_xor(v, d, 32));
  return v;
}
__device__ __forceinline__ float wave_sum(float v) {
#pragma unroll
  for (int d = 16; d >= 1; d >>= 1) v += __shfl_xor(v, d, 32);
  return v;
}

// ---- one block per row; S = row length (compile-time) ---------------------
template <int S>
__global__ __launch_bounds__(NTHREADS) void softmax_rows_kernel(
    const float* __restrict__ x, float* __restrict__ out, long long seg_off) {
  constexpr int VPT = S / NTHREADS;  // 2 / 4 / 6 / 8 floats per thread
  constexpr int CHUNKS = S / 4;      // 16-byte chunks per row

  __shared__ float s_row[S];
  __shared__ float s_red[NWAVES];
  __shared__ float s_max, s_sum;

  const int tid  = threadIdx.x;
  const int lane = tid & 31;
  const int wave = tid >> 5;
  const long long base = seg_off + (long long)blockIdx.x * (long long)S;
  const float* grow = x + base;

  // Async-stage the whole row into LDS (one global read of each byte).
  for (int c = tid; c < CHUNKS; c += NTHREADS)
    async_copy_b128(grow + 4 * c, &s_row[4 * c]);
  wait_async0();     // my wave's async loads complete
  __syncthreads();   // all waves' loads complete

  // Thread-local slice (stride-NTHREADS: conflict-free LDS, coalesced store).
  float v[VPT];
  float m = -3.402823466e38f;
#pragma unroll
  for (int k = 0; k < VPT; ++k) {
    v[k] = s_row[tid + k * NTHREADS];
    m = fmaxf(m, v[k]);
  }

  // Row max: wave butterfly + cross-wave LDS tree.
  m = wave_max(m);
  if (lane == 0) s_red[wave] = m;
  __syncthreads();
  if (wave == 0) {
    float t = (lane < NWAVES) ? s_red[lane] : -3.402823466e38f;
    t = wave_max(t);
    if (lane == 0) s_max = t;
  }
  __syncthreads();
  const float rmax = s_max;

  // exp + row sum.
  float psum = 0.0f;
#pragma unroll
  for (int k = 0; k < VPT; ++k) {
    v[k] = __expf(v[k] - rmax);
    psum += v[k];
  }
  psum = wave_sum(psum);
  if (lane == 0) s_red[wave] = psum;
  __syncthreads();
  if (wave == 0) {
    float t = (lane < NWAVES) ? s_red[lane] : 0.0f;
    t = wave_sum(t);
    if (lane == 0) s_sum = t;
  }
  __syncthreads();
  const float inv = 1.0f / s_sum;

  // Normalize + streaming (non-temporal) store: output never re-read.
  float* orow = out + base;
#pragma unroll
  for (int k = 0; k < VPT; ++k)
    __builtin_nontemporal_store(v[k] * inv, orow + tid + k * NTHREADS);
}

// ---- host launcher ---------------------------------------------------------
extern "C" void kernel_launch(void* const* d_in, const int* in_sizes, int n_in,
                              void* d_out, int out_size, void* d_ws, size_t ws_size,
                              hipStream_t stream) {
  (void)in_sizes; (void)n_in; (void)out_size; (void)d_ws; (void)ws_size;
  const float* x = (const float*)d_in[0];
  float* out = (float*)d_out;
  const int H = 16;  // HEAD_NUM (static in reference)

  long long off = 0;
  // SEQ_LEN = (1024, 2048, 512, 1536): rows per segment = H * s, row len = s.
  softmax_rows_kernel<1024><<<dim3(H * 1024), dim3(NTHREADS), 0, stream>>>(x, out, off);
  off += (long long)H * 1024 * 1024;
  softmax_rows_kernel<2048><<<dim3(H * 2048), dim3(NTHREADS), 0, stream>>>(x, out, off);
  off += (long long)H * 2048 * 2048;
  softmax_rows_kernel<512><<<dim3(H * 512), dim3(NTHREADS), 0, stream>>>(x, out, off);
  off += (long long)H * 512 * 512;
  softmax_rows_kernel<1536><<<dim3(H * 1536), dim3(NTHREADS), 0, stream>>>(x, out, off);
}